// LINK_62689342652831
// MI455X (gfx1250) — compile-verified
//
#include <hip/hip_runtime.h>
#include <hip/hip_bf16.h>
#include <climits>

#define CHANS 64
#define WAVES_PER_BLOCK 8
#define BATCH 8     // edges per async batch (8 x 256B rows in flight per buffer)
#define TJ 128      // TDM transpose tile width (columns of W per block)

#if defined(__AMDGCN__)
#if __has_builtin(__builtin_amdgcn_tensor_load_to_lds)
#define HAVE_TDM 1
#endif
#endif

typedef unsigned int v4u __attribute__((ext_vector_type(4)));
typedef int          v8i __attribute__((ext_vector_type(8)));
typedef int          v4i __attribute__((ext_vector_type(4)));

// ---------------------------------------------------------------------------
// TDM-based tile transpose: W [64, N] -> Wtr [N, 64], full 128-col tiles.
// One TENSOR_LOAD_TO_LDS per block DMAs a 64x128 f32 tile (rows strided by N)
// into LDS with hardware padding (+1 DWORD per 128) so transposed reads are
// bank-conflict-free (stride 129 % 64 == 1). Waves then store Wtr coalesced.
// ---------------------------------------------------------------------------
__global__ void tdm_transpose_kernel(const float* __restrict__ W,
                                     float* __restrict__ Wtr, int N) {
    __shared__ float tile[CHANS * (TJ + 1)];   // 64 rows x 129 dwords
    const int j0 = blockIdx.x * TJ;            // full tiles only: j0+TJ <= N

#ifdef HAVE_TDM
    if (threadIdx.x < 32) {
        unsigned lds_base = (unsigned)(unsigned long long)(&tile[0]);
        unsigned long long ga =
            (unsigned long long)W + (unsigned long long)j0 * 4ull;
        unsigned uN = (unsigned)N;

        // D# group 0: count=1 | lds_addr | global_addr[56:0] | type=2
        v4u g0 = { 1u,
                   lds_base,
                   (unsigned)(ga & 0xFFFFFFFFull),
                   (unsigned)((ga >> 32) & 0x01FFFFFFull) | 0x80000000u };

        // D# group 1:
        //  w0: data_size=2(4B)<<16 | pad_enable<<20 | pad_interval=6<<22
        //  tensor_dim0=N [79:48], tensor_dim1=64 [111:80],
        //  tile_dim0=128 [127:112], tile_dim1=64 [143:128],
        //  tensor_dim0_stride=N [207:160]
        v8i g1 = { (int)((2u << 16) | (1u << 20) | (6u << 22)),
                   (int)((uN & 0xFFFFu) << 16),
                   (int)(((uN >> 16) & 0xFFFFu) | (64u << 16)),
                   (int)(((unsigned)TJ) << 16),
                   (int)64,
                   (int)uN,
                   0, 0 };
        v4i g2 = { 0, 0, 0, 0 };   // <=2D tensor: groups 2/3 unused
        v4i g3 = { 0, 0, 0, 0 };
        v8i gz = { 0, 0, 0, 0, 0, 0, 0, 0 };   // extra arg of 6-arg builtin

        __builtin_amdgcn_tensor_load_to_lds(g0, g1, g2, g3, gz, 0);
        asm volatile("s_wait_tensorcnt 0x0" ::: "memory");
    }
    __syncthreads();
    for (int k = threadIdx.x; k < TJ * CHANS; k += 256) {
        int c  = k & (CHANS - 1);
        int jj = k >> 6;
        Wtr[(unsigned long long)(j0 + jj) * CHANS + c] = tile[c * (TJ + 1) + jj];
    }
#else
    // Fallback: direct strided copy (no TDM builtin on this toolchain).
    for (int k = threadIdx.x; k < TJ * CHANS; k += 256) {
        int c  = k & (CHANS - 1);
        int jj = k >> 6;
        Wtr[(unsigned long long)(j0 + jj) * CHANS + c] =
            W[(unsigned long long)c * N + (j0 + jj)];
    }
#endif
}

// ---------------------------------------------------------------------------
// Scatter-add with CDNA5 async global->LDS gather pipeline (ASYNCcnt).
// One wave32 per batch of 8 edges; double-buffered GLOBAL_LOAD_ASYNC_TO_LDS_B64
// row gathers, s_wait_asynccnt 8/0, readlane-broadcast metadata, fp32 atomics.
// ---------------------------------------------------------------------------
__global__ void scatter_kernel(const int* __restrict__ rows,
                               const int* __restrict__ cols,
                               const float* __restrict__ Wtr,   // [N][64]
                               float* __restrict__ out,
                               const int* __restrict__ minp,
                               int E, int nbatches) {
    __shared__ float lds[WAVES_PER_BLOCK][2][BATCH][CHANS];
    const int lane = threadIdx.x & 31;
    const int wv   = threadIdx.x >> 5;
    const int wave = (blockIdx.x * blockDim.x + threadIdx.x) >> 5;
    const int nwaves = (gridDim.x * blockDim.x) >> 5;
    const int minr = *minp;

    const unsigned lds_base =
        (unsigned)(unsigned long long)(&lds[wv][0][0][0]);

    int b = wave;
    if (b < nbatches) {
        int e0 = b * BATCH;
        int cv0 = cols[e0 + (lane & 7)];
        int rv0 = rows[e0 + (lane & 7)];
        #pragma unroll
        for (int d = 0; d < BATCH; ++d) {
            int c = __builtin_amdgcn_readlane(cv0, d);
            unsigned la = lds_base + d * 256u + lane * 8u;
            const float* g = Wtr + (unsigned long long)c * CHANS + lane * 2;
            asm volatile("global_load_async_to_lds_b64 %0, %1, off"
                         :: "v"(la), "v"(g) : "memory");
        }
        int p = 0;
        int bn = b + nwaves;
        while (bn < nbatches) {
            int e1 = bn * BATCH;
            int cv1 = cols[e1 + (lane & 7)];
            int rv1 = rows[e1 + (lane & 7)];
            int q = p ^ 1;
            #pragma unroll
            for (int d = 0; d < BATCH; ++d) {
                int c = __builtin_amdgcn_readlane(cv1, d);
                unsigned la = lds_base + q * 2048u + d * 256u + lane * 8u;
                const float* g = Wtr + (unsigned long long)c * CHANS + lane * 2;
                asm volatile("global_load_async_to_lds_b64 %0, %1, off"
                             :: "v"(la), "v"(g) : "memory");
            }
            asm volatile("s_wait_asynccnt 0x8" ::: "memory");
            #pragma unroll
            for (int d = 0; d < BATCH; ++d) {
                int r = __builtin_amdgcn_readlane(rv0, d) - minr;
                float2 w = *(const float2*)&lds[wv][p][d][lane * 2];
                float* dst = out + (unsigned long long)r * CHANS + lane * 2;
                __hip_atomic_fetch_add(dst,     w.x, __ATOMIC_RELAXED, __HIP_MEMORY_SCOPE_AGENT);
                __hip_atomic_fetch_add(dst + 1, w.y, __ATOMIC_RELAXED, __HIP_MEMORY_SCOPE_AGENT);
            }
            rv0 = rv1;
            p = q;
            bn += nwaves;
        }
        asm volatile("s_wait_asynccnt 0x0" ::: "memory");
        #pragma unroll
        for (int d = 0; d < BATCH; ++d) {
            int r = __builtin_amdgcn_readlane(rv0, d) - minr;
            float2 w = *(const float2*)&lds[wv][p][d][lane * 2];
            float* dst = out + (unsigned long long)r * CHANS + lane * 2;
            __hip_atomic_fetch_add(dst,     w.x, __ATOMIC_RELAXED, __HIP_MEMORY_SCOPE_AGENT);
            __hip_atomic_fetch_add(dst + 1, w.y, __ATOMIC_RELAXED, __HIP_MEMORY_SCOPE_AGENT);
        }
    }

    // tail edges (E not multiple of BATCH)
    for (int e = nbatches * BATCH + wave; e < E; e += nwaves) {
        int r = rows[e] - minr;
        int c = cols[e];
        __builtin_prefetch(Wtr + (unsigned long long)c * CHANS, 0, 3);
        float2 w = *(const float2*)(Wtr + (unsigned long long)c * CHANS + lane * 2);
        float* dst = out + (unsigned long long)r * CHANS + lane * 2;
        __hip_atomic_fetch_add(dst,     w.x, __ATOMIC_RELAXED, __HIP_MEMORY_SCOPE_AGENT);
        __hip_atomic_fetch_add(dst + 1, w.y, __ATOMIC_RELAXED, __HIP_MEMORY_SCOPE_AGENT);
    }
}

// ---------------------------------------------------------------------------
// Zero the output accumulator + init row-min slot.
// ---------------------------------------------------------------------------
__global__ void zero_init_kernel(float4* out4, int n4, int* minp) {
    int t = blockIdx.x * blockDim.x + threadIdx.x;
    if (t == 0) *minp = INT_MAX;
    for (int i = t; i < n4; i += gridDim.x * blockDim.x) {
        out4[i] = make_float4(0.f, 0.f, 0.f, 0.f);
    }
}

// ---------------------------------------------------------------------------
// Min over edge_index[0]  (reference: row = row - row.min())
// ---------------------------------------------------------------------------
__global__ void row_min_kernel(const int* __restrict__ rows, int E, int* minp) {
    __shared__ int smin[256];
    int t = blockIdx.x * blockDim.x + threadIdx.x;
    int lmin = INT_MAX;
    for (int i = t; i < E; i += gridDim.x * blockDim.x) {
        int v = rows[i];
        lmin = v < lmin ? v : lmin;
    }
    smin[threadIdx.x] = lmin;
    __syncthreads();
    for (int off = 128; off > 0; off >>= 1) {
        if (threadIdx.x < off) {
            int o = smin[threadIdx.x + off];
            if (o < smin[threadIdx.x]) smin[threadIdx.x] = o;
        }
        __syncthreads();
    }
    if (threadIdx.x == 0) atomicMin(minp, smin[0]);
}

// ---------------------------------------------------------------------------
// Remainder-column transpose (columns not covered by full TDM tiles).
// ---------------------------------------------------------------------------
__global__ void transpose_kernel(const float* __restrict__ W,
                                 float* __restrict__ Wtr, int N, int jstart) {
    int j = jstart + blockIdx.x * blockDim.x + threadIdx.x;
    int c = blockIdx.y;
    if (j < N) {
        Wtr[(long long)j * CHANS + c] = W[(long long)c * N + j];
    }
}

// ---------------------------------------------------------------------------
// Bias + log-softmax: one wave32 per 64-ch row; lane owns ch 2L, 2L+1 so all
// global traffic is b64. Wave32 shuffle-xor butterflies for max and sum.
// ---------------------------------------------------------------------------
__global__ void logsoftmax_kernel(float* __restrict__ out,
                                  const float* __restrict__ bias, int N) {
    int lane = threadIdx.x & 31;
    int row = (blockIdx.x * blockDim.x + threadIdx.x) >> 5;
    if (row >= N) return;  // whole wave exits together

    float2* rowp2 = (float2*)(out + (long long)row * CHANS);
    const float2* bias2 = (const float2*)bias;
    float2 xv = rowp2[lane];
    float2 bv = bias2[lane];
    float x0 = xv.x + bv.x;
    float x1 = xv.y + bv.y;

    float m = fmaxf(x0, x1);
    #pragma unroll
    for (int off = 16; off > 0; off >>= 1)
        m = fmaxf(m, __shfl_xor(m, off, 32));

    float s = __expf(x0 - m) + __expf(x1 - m);
    #pragma unroll
    for (int off = 16; off > 0; off >>= 1)
        s += __shfl_xor(s, off, 32);

    float lse = m + __logf(s);
    rowp2[lane] = make_float2(x0 - lse, x1 - lse);
}

// ---------------------------------------------------------------------------
// Host-side launcher
// ---------------------------------------------------------------------------
extern "C" void kernel_launch(void* const* d_in, const int* in_sizes, int n_in,
                              void* d_out, int out_size, void* d_ws, size_t ws_size,
                              hipStream_t stream) {
    const int* edge_index = (const int*)d_in[0];   // [2, E] flat
    const float* W        = (const float*)d_in[1]; // [64, N]
    const float* bias     = (const float*)d_in[2]; // [64]

    const int E = in_sizes[0] / 2;
    const int N = in_sizes[1] / CHANS;

    const int* rows = edge_index;
    const int* cols = edge_index + E;

    int* minp  = (int*)d_ws;
    float* Wtr = (float*)((char*)d_ws + 256);
    float* out = (float*)d_out;

    // 1) zero accumulator + init min
    zero_init_kernel<<<4096, 256, 0, stream>>>((float4*)out, (N * CHANS) / 4, minp);
    // 2) row min
    row_min_kernel<<<1024, 256, 0, stream>>>(rows, E, minp);
    // 3) transpose W -> Wtr: TDM full tiles + elementwise remainder
    {
        int full = N / TJ;
        if (full > 0)
            tdm_transpose_kernel<<<full, 256, 0, stream>>>(W, Wtr, N);
        int jrem = full * TJ;
        if (jrem < N)
            transpose_kernel<<<dim3((N - jrem + 255) / 256, CHANS), 256, 0, stream>>>(
                W, Wtr, N, jrem);
    }
    // 4) async-pipelined scatter-add
    scatter_kernel<<<8192, 256, 0, stream>>>(rows, cols, Wtr, out, minp,
                                             E, E / BATCH);
    // 5) bias + log-softmax (8 rows per 256-thread block)
    logsoftmax_kernel<<<(N + 7) / 8, 256, 0, stream>>>(out, bias, N);
}